// R_norm_6914897347085
// MI455X (gfx1250) — compile-verified
//
#include <hip/hip_runtime.h>
#include <hip/hip_bf16.h>

#define NUM_CLASSES 13
#define CLEN (NUM_CLASSES * NUM_CLASSES)   // 169 floats per C matrix
#define ROWS 64                            // rows (matrices) per block
#define BLOCK 64                           // threads per block (2 waves)
#define TILE_F4 (ROWS * CLEN / 4)          // 2704 float4 per C tile (16B-aligned)
#define FBLOCK 256                         // final-reduce block size

#define AS1 __attribute__((address_space(1)))
#define AS3 __attribute__((address_space(3)))

#if defined(__gfx1250__)
#  if __has_builtin(__builtin_amdgcn_global_load_async_to_lds_b128)
#    define HAVE_ASYNC_LDS 1
#  endif
#endif

typedef __attribute__((ext_vector_type(2))) float v2f;
typedef __attribute__((ext_vector_type(4))) float fl4;
typedef __attribute__((ext_vector_type(8))) float v8f;
typedef int v4i __attribute__((vector_size(16)));   // matches builtin param pointee

// Exact 32-lane (wave32) sum reduction using V_WMMA_F32_16X16X4_F32.
// A (16x4 f32): VGPR0 -> lanes 0-15 = (M=lane,K=0), lanes 16-31 = (M=lane-16,K=2);
// VGPR1 (K=1/K=3) zeroed. B = ones -> D[m][*] = val(m)+val(m+16). Summing the 8
// D VGPRs gives low/high half-sums; shfl_xor(16) completes. Exact in f32 (B=1, RNE).
__device__ __forceinline__ float wave32_sum_wmma(float v) {
    v2f a; a.x = v;    a.y = 0.0f;
    v2f b; b.x = 1.0f; b.y = 1.0f;
    v8f c = {};
    v8f d = __builtin_amdgcn_wmma_f32_16x16x4_f32(
        false, a, false, b, (short)0, c, false, false);
    float s = d[0] + d[1] + d[2] + d[3] + d[4] + d[5] + d[6] + d[7];
    s += __shfl_xor(s, 16, 32);
    return s;
}

// Block-level: stage the block's contiguous C region (ROWS*676 B) into LDS with
// coalesced 16B transfers (async global->LDS on gfx1250), then each thread does
// the 13x13 matvec + norms from its conflict-free LDS row.
__global__ __launch_bounds__(BLOCK) void rnorm_main_kernel(
    const float* __restrict__ inp,     // [N,13]
    const int*   __restrict__ target,  // [N]
    const float* __restrict__ s,       // [N,13]
    const float* __restrict__ Cm,      // [N,13,13]
    float* __restrict__ block_partials,
    int N)
{
    __shared__ __align__(16) float Cs[ROWS * CLEN];   // 43,264 B
    __shared__ float red[BLOCK];

    const int tx   = threadIdx.x;
    const int base = blockIdx.x * ROWS;
    const int tid  = base + tx;

    const float* Cg = Cm + (size_t)base * CLEN;

    if (base + ROWS <= N) {
        // Full tile: 2704 coalesced 16B transfers, striped across 64 threads.
#if defined(HAVE_ASYNC_LDS)
        AS1 v4i* gsrc = (AS1 v4i*)Cg;   // global source (addrspace 1)
        AS3 v4i* ldst = (AS3 v4i*)Cs;   // LDS destination (addrspace 3)
        for (int i = tx; i < TILE_F4; i += BLOCK) {
            __builtin_amdgcn_global_load_async_to_lds_b128(gsrc + i, ldst + i, 0, 0);
        }
        asm volatile("s_wait_asynccnt 0x0" ::: "memory");
#else
        const fl4* __restrict__ g4 = (const fl4*)Cg;
        fl4* l4 = (fl4*)Cs;
        for (int i = tx; i < TILE_F4; i += BLOCK)
            l4[i] = g4[i];
#endif
    } else {
        // Ragged tail tile: scalar staging with bounds guard.
        const int nf = (N > base ? (N - base) : 0) * CLEN;
        for (int i = tx; i < nf; i += BLOCK)
            Cs[i] = Cg[i];
    }
    __syncthreads();

    float acc = 0.0f;
    if (tid < N) {
        const float* myC = &Cs[tx * CLEN];   // LDS lane stride 169 -> conflict-free
        const float* sn  = s   + (size_t)tid * NUM_CLASSES;
        const float* xn  = inp + (size_t)tid * NUM_CLASSES;
        const int t = target[tid];

        float x[NUM_CLASSES], sv[NUM_CLASSES];
#pragma unroll
        for (int j = 0; j < NUM_CLASSES; ++j) {
            x[j]  = xn[j];
            sv[j] = sn[j];
        }

        float zp2 = 0.0f, zy2 = 0.0f, dg = 0.0f;
#pragma unroll
        for (int i = 0; i < NUM_CLASSES; ++i) {
            const float* row = myC + i * NUM_CLASSES;
            float ye = 0.0f;
#pragma unroll
            for (int j = 0; j < NUM_CLASSES; ++j)
                ye = fmaf(row[j], x[j], ye);
            const float zp = sv[i] - row[t];   // s - C[:,target]
            const float zy = sv[i] - ye;       // s - C @ input
            zp2 = fmaf(zp, zp, zp2);
            zy2 = fmaf(zy, zy, zy2);
            dg  = fmaf(zy, zp, dg);
        }
        acc = sqrtf(zp2) - dg / sqrtf(zy2);
    }

    // Deterministic in-block reduction: 64 -> 32 via LDS, then WMMA wave reduce.
    red[tx] = acc;
    __syncthreads();
    if (tx < 32) red[tx] += red[tx + 32];
    __syncthreads();
    if (tx < 32) {   // wave 0, EXEC all-1s -> WMMA legal
        const float total = wave32_sum_wmma(red[tx]);
        if (tx == 0) block_partials[blockIdx.x] = total;
    }
}

// Single-block deterministic final reduction of block partials -> d_out[0].
__global__ __launch_bounds__(FBLOCK) void rnorm_final_kernel(
    const float* __restrict__ partials, int nparts, float* __restrict__ out)
{
    __shared__ float red[FBLOCK];
    const int tx = threadIdx.x;
    float a = 0.0f;
    for (int i = tx; i < nparts; i += FBLOCK)   // fixed order per thread
        a += partials[i];
    red[tx] = a;
    __syncthreads();
#pragma unroll
    for (int off = FBLOCK / 2; off >= 32; off >>= 1) {
        if (tx < off) red[tx] += red[tx + off];
        __syncthreads();
    }
    if (tx < 32) {
        const float total = wave32_sum_wmma(red[tx]);
        if (tx == 0) out[0] = total;
    }
}

extern "C" void kernel_launch(void* const* d_in, const int* in_sizes, int n_in,
                              void* d_out, int out_size, void* d_ws, size_t ws_size,
                              hipStream_t stream) {
    // setup_inputs order: input[N,13] f32, target[N] int, s[N,13] f32,
    //                     C_matrix[N,13,13] f32, instance_weights[N] f32 (unused)
    const float* inp = (const float*)d_in[0];
    const int*   tgt = (const int*)d_in[1];
    const float* s   = (const float*)d_in[2];
    const float* Cm  = (const float*)d_in[3];
    (void)n_in; (void)ws_size; (void)out_size;

    const int N = in_sizes[1];                  // element count of target == N
    const int blocks = (N + ROWS - 1) / ROWS;   // 4096 for N = 262144

    float* partials = (float*)d_ws;             // blocks * 4 B of scratch (16 KB)

    rnorm_main_kernel<<<blocks, BLOCK, 0, stream>>>(inp, tgt, s, Cm, partials, N);
    rnorm_final_kernel<<<1, FBLOCK, 0, stream>>>(partials, blocks, (float*)d_out);
}